// PointNet_SA_Module_26972394619815
// MI455X (gfx1250) — compile-verified
//
#include <hip/hip_runtime.h>

// Problem constants (match reference setup_inputs)
#define BB 16
#define NN 4096
#define MM 1024
#define CC 64
#define KK 32
#define BM (BB*MM)          // 16384 query points
#define CNT_F 524288.0f     // B*M*K elements per channel for BN stats

typedef __attribute__((ext_vector_type(2))) float v2f;
typedef __attribute__((ext_vector_type(8))) float v8f;

__device__ __forceinline__ v8f wmma_f32(v2f a, v2f b, v8f c) {
  // CDNA5 fp32 matrix op: D(16x16) = A(16x4) * B(4x16) + C
  return __builtin_amdgcn_wmma_f32_16x16x4_f32(false, a, false, b, (short)0, c, false, false);
}

// ---------------------------------------------------------------------------
// K0: zero the BN stat accumulators (512 floats) — must happen every launch
// ---------------------------------------------------------------------------
__global__ void init_stats_kernel(float* __restrict__ st) {
  st[threadIdx.x] = 0.0f;   // launched with 512 threads
}

// ---------------------------------------------------------------------------
// K1: ball query — one wave per query point, ordered ballot-append of indices
// ---------------------------------------------------------------------------
__global__ void __launch_bounds__(256)
ball_query_kernel(const float* __restrict__ xyz, const float* __restrict__ new_xyz,
                  int* __restrict__ inds) {
  const int lane = threadIdx.x & 31;
  const int g = blockIdx.x * 8 + (threadIdx.x >> 5);   // query id in [0, B*M)
  const int b = g >> 10;
  const int m = g & (MM - 1);
  const float* q = new_xyz + (size_t)(b * MM + m) * 3;
  const float qx = q[0], qy = q[1], qz = q[2];
  const float qq = qx*qx + qy*qy + qz*qz;
  int* out = inds + (size_t)g * KK;
  int cnt = 0;
  int first = NN - 1;
  for (int n0 = 0; n0 < NN && cnt < KK; n0 += 32) {
    const int n = n0 + lane;
    const float* p = xyz + ((size_t)b * NN + n) * 3;
    const float px = p[0], py = p[1], pz = p[2];
    const float pp = px*px + py*py + pz*pz;
    // same algebraic form as reference (qq + pp - 2 q.p) for matching rounding
    const float d2 = qq + pp - 2.0f * (qx*px + qy*py + qz*pz);
    const bool in = (d2 <= 0.25f);
    const unsigned msk = (unsigned)__ballot(in);
    if (msk) {
      if (cnt == 0) first = n0 + __builtin_ctz(msk);
      const int pos = cnt + __popc(msk & ((1u << lane) - 1u));
      if (in && pos < KK) out[pos] = n;
      cnt += __popc(msk);
    }
  }
  if (cnt > KK) cnt = KK;
  for (int k = cnt + lane; k < KK; k += 32) out[k] = first;   // pad with first hit
}

// ---------------------------------------------------------------------------
// K2: gather + layer0 GEMM (32x67 @ 67x64) via V_WMMA_F32_16X16X4_F32
//     2 waves / block, one query point per wave; LDS pre-reduced BN stats
// ---------------------------------------------------------------------------
__global__ void __launch_bounds__(64)
layer0_kernel(const float* __restrict__ xyz, const float* __restrict__ points,
              const float* __restrict__ W0, const int* __restrict__ inds,
              float* __restrict__ y0, float* __restrict__ gsum, float* __restrict__ gsq) {
  __shared__ float Wl[64 * 68];       // 64x67 weights padded to stride 68 (+zero col)
  __shared__ float Xl[2][32 * 68];    // per-wave gathered features, stride 68
  __shared__ float lst[128];          // block-local [sum(64) | sumsq(64)]
  const int tid = threadIdx.x, lane = tid & 31, wv = tid >> 5;
  const int r0 = lane & 15;
  const int cb_hi = (lane >> 4) << 1;

  for (int i = tid; i < 64 * 68; i += 64) {
    const int r = i / 68, c = i % 68;
    Wl[i] = (c < 67) ? W0[r * 67 + c] : 0.0f;
  }
  for (int i = tid; i < 128; i += 64) lst[i] = 0.0f;

  const int g = blockIdx.x * 2 + wv;
  const int b = g >> 10;
  const int myind = inds[(size_t)g * KK + lane];
  float* X = Xl[wv];
  for (int k = 0; k < KK; ++k) {
    const int row = __shfl(myind, k, 32);
    const float* px = xyz + ((size_t)b * NN + row) * 3;
    const float* pf = points + ((size_t)b * NN + row) * CC;
    if (lane < 3) X[k * 68 + lane] = px[lane];
    X[k * 68 + 3 + lane]  = pf[lane];
    X[k * 68 + 35 + lane] = pf[32 + lane];
    if (lane == 0) X[k * 68 + 67] = 0.0f;   // zero pad col (c==67)
  }
  __syncthreads();

  v8f acc[2][4] = {};
  for (int kk = 0; kk < 17; ++kk) {         // ceil(67/4) K-steps, last uses zero pad
    const int cb = kk * 4 + cb_hi;
    v2f a0, a1;
    a0[0] = X[r0 * 68 + cb];        a0[1] = X[r0 * 68 + cb + 1];
    a1[0] = X[(r0 + 16) * 68 + cb]; a1[1] = X[(r0 + 16) * 68 + cb + 1];
#pragma unroll
    for (int nt = 0; nt < 4; ++nt) {
      v2f bf;
      const int o = nt * 16 + r0;
      bf[0] = Wl[o * 68 + cb]; bf[1] = Wl[o * 68 + cb + 1];
      acc[0][nt] = wmma_f32(a0, bf, acc[0][nt]);
      acc[1][nt] = wmma_f32(a1, bf, acc[1][nt]);
    }
  }

  float* yo = y0 + (size_t)g * (KK * 64);
#pragma unroll
  for (int nt = 0; nt < 4; ++nt) {
    float s = 0.0f, sq = 0.0f;
    const int col = nt * 16 + r0;
#pragma unroll
    for (int mt = 0; mt < 2; ++mt) {
#pragma unroll
      for (int r = 0; r < 8; ++r) {
        const int row = mt * 16 + ((lane >> 4) << 3) + r;
        const float v = acc[mt][nt][r];
        yo[row * 64 + col] = v;
        s += v; sq += v * v;
      }
    }
    atomicAdd(&lst[col], s);
    atomicAdd(&lst[64 + col], sq);
  }
  __syncthreads();
  if (tid < 64) {
    atomicAdd(&gsum[tid], lst[tid]);
    atomicAdd(&gsq[tid],  lst[64 + tid]);
  }
}

// ---------------------------------------------------------------------------
// K4/K6: BN+ReLU(previous) then GEMM with W (O x 64). O=64 stores y; O=128
//        skips storing y2 and instead records per-(b,m) channel max & min.
// ---------------------------------------------------------------------------
template <int O, int NT>
__global__ void __launch_bounds__(64)
layerN_kernel(const float* __restrict__ yin, const float* __restrict__ W,
              const float* __restrict__ scale, const float* __restrict__ shift,
              float* __restrict__ yout, float* __restrict__ gsum, float* __restrict__ gsq,
              float* __restrict__ mm) {
  __shared__ float Wl[O * 68];
  __shared__ float Xl[2][32 * 68];
  __shared__ float lst[2 * O];
  const int tid = threadIdx.x, lane = tid & 31, wv = tid >> 5;
  const int r0 = lane & 15;
  const int cb_hi = (lane >> 4) << 1;

  for (int i = tid; i < O * 68; i += 64) {
    const int r = i / 68, c = i % 68;
    Wl[i] = (c < 64) ? W[r * 64 + c] : 0.0f;
  }
  for (int i = tid; i < 2 * O; i += 64) lst[i] = 0.0f;

  const int g = blockIdx.x * 2 + wv;
  const float* yi = yin + (size_t)g * (KK * 64);
  float* X = Xl[wv];
  const float sc0 = scale[lane],      sh0 = shift[lane];
  const float sc1 = scale[lane + 32], sh1 = shift[lane + 32];
  for (int k = 0; k < KK; ++k) {
    X[k * 68 + lane]      = fmaxf(yi[k * 64 + lane]      * sc0 + sh0, 0.0f);
    X[k * 68 + lane + 32] = fmaxf(yi[k * 64 + lane + 32] * sc1 + sh1, 0.0f);
  }
  __syncthreads();

  v8f acc[2][NT] = {};
  for (int kk = 0; kk < 16; ++kk) {
    const int cb = kk * 4 + cb_hi;
    v2f a0, a1;
    a0[0] = X[r0 * 68 + cb];        a0[1] = X[r0 * 68 + cb + 1];
    a1[0] = X[(r0 + 16) * 68 + cb]; a1[1] = X[(r0 + 16) * 68 + cb + 1];
#pragma unroll
    for (int nt = 0; nt < NT; ++nt) {
      v2f bf;
      const int o = nt * 16 + r0;
      bf[0] = Wl[o * 68 + cb]; bf[1] = Wl[o * 68 + cb + 1];
      acc[0][nt] = wmma_f32(a0, bf, acc[0][nt]);
      acc[1][nt] = wmma_f32(a1, bf, acc[1][nt]);
    }
  }

  float* yo = yout ? (yout + (size_t)g * (KK * 64)) : nullptr;
#pragma unroll
  for (int nt = 0; nt < NT; ++nt) {
    float s = 0.0f, sq = 0.0f, mx = -3.4e38f, mn = 3.4e38f;
    const int col = nt * 16 + r0;
#pragma unroll
    for (int mt = 0; mt < 2; ++mt) {
#pragma unroll
      for (int r = 0; r < 8; ++r) {
        const float v = acc[mt][nt][r];
        if (yo) {
          const int row = mt * 16 + ((lane >> 4) << 3) + r;
          yo[row * 64 + col] = v;
        }
        s += v; sq += v * v;
        mx = fmaxf(mx, v); mn = fminf(mn, v);
      }
    }
    atomicAdd(&lst[col], s);
    atomicAdd(&lst[O + col], sq);
    if (mm) {   // fused pooling pre-reduction: raw max & min per (b,m,channel)
      mx = fmaxf(mx, __shfl_xor(mx, 16, 32));
      mn = fminf(mn, __shfl_xor(mn, 16, 32));
      if (lane < 16) {
        float* row = mm + (size_t)g * (2 * O);
        row[col] = mx; row[O + col] = mn;
      }
    }
  }
  __syncthreads();
  for (int i = tid; i < O; i += 64) {
    atomicAdd(&gsum[i], lst[i]);
    atomicAdd(&gsq[i],  lst[O + i]);
  }
}

// ---------------------------------------------------------------------------
// K3/K5/K7: finalize BN stats -> fused per-channel scale/shift
// ---------------------------------------------------------------------------
__global__ void finalize_kernel(const float* __restrict__ gsum, const float* __restrict__ gsq,
                                const float* __restrict__ gg, const float* __restrict__ bb,
                                float* __restrict__ scale, float* __restrict__ shift, int O) {
  const int t = threadIdx.x;
  if (t < O) {
    const float mu  = gsum[t] / CNT_F;
    const float var = gsq[t] / CNT_F - mu * mu;
    const float rs  = rsqrtf(var + 1e-5f);
    const float sc  = rs * gg[t];
    scale[t] = sc;
    shift[t] = bb[t] - mu * sc;
  }
}

// ---------------------------------------------------------------------------
// K8: copy new_xyz into output head
// ---------------------------------------------------------------------------
__global__ void copy_newxyz_kernel(const float* __restrict__ src, float* __restrict__ dst) {
  const int i = blockIdx.x * 256 + threadIdx.x;
  dst[i] = src[i];
}

// ---------------------------------------------------------------------------
// K9: pooled output. BN affine + ReLU commute with max/min pre-reduction:
//     pooled = relu(scale*max + shift) if scale>=0 else relu(scale*min + shift)
// ---------------------------------------------------------------------------
__global__ void __launch_bounds__(256)
pool_kernel(const float* __restrict__ mm, const float* __restrict__ scale,
            const float* __restrict__ shift, float* __restrict__ out) {
  const int idx = blockIdx.x * 256 + threadIdx.x;   // [0, B*M*128)
  const int c  = idx & 127;
  const int bm = idx >> 7;
  const float s = scale[c], sh = shift[c];
  const float mx = mm[(size_t)bm * 256 + c];
  const float mn = mm[(size_t)bm * 256 + 128 + c];
  const float v = (s >= 0.0f) ? mx : mn;
  out[idx] = fmaxf(v * s + sh, 0.0f);
}

// ---------------------------------------------------------------------------
extern "C" void kernel_launch(void* const* d_in, const int* in_sizes, int n_in,
                              void* d_out, int out_size, void* d_ws, size_t ws_size,
                              hipStream_t stream) {
  (void)in_sizes; (void)n_in; (void)out_size; (void)ws_size;
  const float* xyz     = (const float*)d_in[0];
  const float* points  = (const float*)d_in[1];
  const float* new_xyz = (const float*)d_in[2];
  const float* W0 = (const float*)d_in[3];
  const float* g0 = (const float*)d_in[4];
  const float* b0 = (const float*)d_in[5];
  const float* W1 = (const float*)d_in[6];
  const float* g1 = (const float*)d_in[7];
  const float* b1 = (const float*)d_in[8];
  const float* W2 = (const float*)d_in[9];
  const float* g2 = (const float*)d_in[10];
  const float* b2 = (const float*)d_in[11];

  float* out = (float*)d_out;
  float* ws  = (float*)d_ws;
  int*   inds = (int*)d_ws;                 // B*M*K ints            (2 MB)
  float* y0 = ws + 524288;                  // B*M*K*64 f32          (134 MB)
  float* y1 = y0 + 33554432;                // B*M*K*64 f32          (134 MB)
  float* mm = y1 + 33554432;                // B*M*(128 max+128 min) (16 MB)
  float* st = mm + 4194304;                 // stats block (1024 floats)
  float* gsum0 = st;        float* gsq0 = st + 64;
  float* gsum1 = st + 128;  float* gsq1 = st + 192;
  float* gsum2 = st + 256;  float* gsq2 = st + 384;
  float* sc0 = st + 512;  float* sh0 = st + 576;
  float* sc1 = st + 640;  float* sh1 = st + 704;
  float* sc2 = st + 768;  float* sh2 = st + 896;

  init_stats_kernel<<<1, 512, 0, stream>>>(st);
  ball_query_kernel<<<BM / 8, 256, 0, stream>>>(xyz, new_xyz, inds);
  layer0_kernel<<<BM / 2, 64, 0, stream>>>(xyz, points, W0, inds, y0, gsum0, gsq0);
  finalize_kernel<<<1, 128, 0, stream>>>(gsum0, gsq0, g0, b0, sc0, sh0, 64);
  layerN_kernel<64, 4><<<BM / 2, 64, 0, stream>>>(y0, W1, sc0, sh0, y1, gsum1, gsq1, nullptr);
  finalize_kernel<<<1, 128, 0, stream>>>(gsum1, gsq1, g1, b1, sc1, sh1, 64);
  layerN_kernel<128, 8><<<BM / 2, 64, 0, stream>>>(y1, W2, sc1, sh1, nullptr, gsum2, gsq2, mm);
  finalize_kernel<<<1, 128, 0, stream>>>(gsum2, gsq2, g2, b2, sc2, sh2, 128);
  copy_newxyz_kernel<<<BM * 3 / 256, 256, 0, stream>>>(new_xyz, out);
  pool_kernel<<<BM * 128 / 256, 256, 0, stream>>>(mm, sc2, sh2, out + BM * 3);
}